// RelativeGlobalAttention_42700564857445
// MI455X (gfx1250) — compile-verified
//
#include <hip/hip_runtime.h>
#include <math.h>

// Relative Global Attention (Music Transformer) for MI455X / gfx1250.
//
// fp32 WMMA (V_WMMA_F32_16X16X4_F32) for all matrix math: the problem is
// HBM-bound on materializing the 268MB attn_weights output, so full fp32
// precision costs nothing at the roofline and keeps the directly-validated
// attn_weights exact.
//
// Key CDNA5-specific optimization: 320KB LDS per WGP lets a whole 16-row
// logits block (16 x 2048 f32 = 128KB) live in LDS, so logits -> softmax ->
// attn@V is fused into ONE kernel. attn_weights touches HBM exactly once
// (the mandatory output write) instead of 4 times.
//
// Shapes: B=2, S=2048, D=512, H=8, hd=64.

#define S 2048
#define D 512
#define H 8
#define HD 64
#define BSZ 2

#define LSTR 2050                      // LDS row stride (floats): %64==2 ->
                                       // conflict-free, 8B-aligned pairs
#define SMEM_FLOATS (16 * LSTR + 1024) // logits block + 4x(16x16) partials
#define SMEM_BYTES  (SMEM_FLOATS * 4)  // 135,296 B -> 2 workgroups per WGP

typedef float v2f __attribute__((ext_vector_type(2)));
typedef float v8f __attribute__((ext_vector_type(8)));

__device__ __forceinline__ v8f wmma4f32(v2f a, v2f b, v8f c) {
  // D(16x16,f32) = A(16x4,f32) * B(4x16,f32) + C
  return __builtin_amdgcn_wmma_f32_16x16x4_f32(
      /*neg_a=*/false, a, /*neg_b=*/false, b,
      /*c_mod=*/(short)0, c, /*reuse_a=*/false, /*reuse_b=*/false);
}

__device__ __forceinline__ v8f vzero8() {
  v8f z = {0.f, 0.f, 0.f, 0.f, 0.f, 0.f, 0.f, 0.f};
  return z;
}

// -------------------------------------------------------------------------
// GEMM: dst = X(4096x512) @ W(512x512) + bias
// out_mode 0: scatter to (B,H,S,hd) head layout; out_mode 1: plain row-major.
// One wave computes a 16x64 output tile (A-fragment reused across 4 N-tiles).
// Grid: 256 blocks x 256 threads = 2048 waves = (4096/16) * (512/64) tiles.
// -------------------------------------------------------------------------
__global__ __launch_bounds__(256) void gemm_proj_kernel(
    const float* __restrict__ X, const float* __restrict__ W,
    const float* __restrict__ bias, float* __restrict__ dst, int out_mode)
{
  const int lane = threadIdx.x & 31;
  const int wid  = blockIdx.x * 8 + (threadIdx.x >> 5);   // 0..2047
  const int m0   = (wid >> 3) << 4;                       // row tile * 16
  const int n0   = (wid & 7) << 6;                        // col tile * 64
  const int half = lane >> 4;
  const int r    = lane & 15;

  v8f acc[4];
#pragma unroll
  for (int j = 0; j < 4; ++j) acc[j] = vzero8();

  const float* arow = X + (size_t)(m0 + r) * D;
#pragma unroll 2
  for (int k = 0; k < D; k += 4) {
    const int ka = k + 2 * half;
    v2f a; a.x = arow[ka]; a.y = arow[ka + 1];
    const float* wp0 = W + (size_t)ka * D + n0 + r;       // row ka
    const float* wp1 = wp0 + D;                           // row ka+1
#pragma unroll
    for (int j = 0; j < 4; ++j) {
      v2f b; b.x = wp0[j * 16]; b.y = wp1[j * 16];
      acc[j] = wmma4f32(a, b, acc[j]);
    }
  }

#pragma unroll
  for (int j = 0; j < 4; ++j) {
    const int n  = n0 + j * 16 + r;
    const float bv = bias[n];
#pragma unroll
    for (int t = 0; t < 8; ++t) {
      const int m = m0 + t + 8 * half;
      const float val = acc[j][t] + bv;
      if (out_mode == 0) {
        const int bb = m >> 11, s = m & (S - 1);
        const int hh = n >> 6,  dc = n & (HD - 1);
        dst[(((size_t)(bb * H + hh)) * S + s) * HD + dc] = val;
      } else {
        dst[(size_t)m * D + n] = val;
      }
    }
  }
}

// -------------------------------------------------------------------------
// Fused attention kernel. One workgroup (8 waves) per (b,h, 16-query block).
//
// Phase 1: logits[q0:q0+16, 0:2048] = (QK^T + skewed QE)/8 -> LDS
//          Skew identity: Srel[q,k] = QE[q, S-1-(q-k)] for k<=q, else 0.
//          Per 16-wide causal k-subtile, two 16x16 QE WMMA tiles covering
//          the diagonal band m0..m0+31 are realigned via __shfl.
// Phase 2: row softmax in LDS; normalized weights written once to d_out and
//          kept in LDS.
// Phase 3: attnV = attn @ V with A-fragments from LDS. 8 waves = 4 n-tiles
//          x 2 K-halves; K-half partials combined via LDS in fixed order
//          (deterministic).
// Grid: BSZ*H*(S/16) = 2048 blocks x 256 threads.
// -------------------------------------------------------------------------
__global__ __launch_bounds__(256) void fused_attn_kernel(
    const float* __restrict__ qh, const float* __restrict__ kh,
    const float* __restrict__ vh, const float* __restrict__ E,
    float* __restrict__ attnW, float* __restrict__ attnV)
{
  extern __shared__ float smem[];                // [16][LSTR] + 1024 partials

  const int lane = threadIdx.x & 31;
  const int w    = threadIdx.x >> 5;             // wave 0..7
  const int qt   = blockIdx.x & 127;
  const int bh   = blockIdx.x >> 7;
  const int q0   = qt << 4;
  const int half = lane >> 4;
  const int r    = lane & 15;

  const float* Q = qh + (size_t)bh * S * HD;
  const float* K = kh + (size_t)bh * S * HD;
  const float* V = vh + (size_t)bh * S * HD;
  const float* qrow = Q + (size_t)(q0 + r) * HD;

  // ======== Phase 1: logits row-block into LDS ========
  for (int c = w; c < 32; c += 8) {              // 4 k64-chunks per wave
    const int k0 = c << 6;

    v8f acc[4];
#pragma unroll
    for (int j = 0; j < 4; ++j) acc[j] = vzero8();

#pragma unroll 4
    for (int d = 0; d < HD; d += 4) {
      const int da = d + 2 * half;
      v2f a; a.x = qrow[da]; a.y = qrow[da + 1];
#pragma unroll
      for (int j = 0; j < 4; ++j) {
        const float* kr = K + (size_t)(k0 + j * 16 + r) * HD + da;
        v2f b; b.x = kr[0]; b.y = kr[1];
        acc[j] = wmma4f32(a, b, acc[j]);
      }
    }

    // relative term per causal k-subtile
#pragma unroll
    for (int j = 0; j < 4; ++j) {
      const int k0j = k0 + j * 16;
      if (k0j <= q0) {
        const int m0 = S - 16 - q0 + k0j;        // >= 0, m0+31 <= S+15
        v8f e0 = vzero8(), e1 = vzero8();
#pragma unroll 4
        for (int d = 0; d < HD; d += 4) {
          const int da = d + 2 * half;
          v2f a; a.x = qrow[da]; a.y = qrow[da + 1];
          const float* er0 = E + (size_t)(m0 + r) * HD + da;
          const float* er1 = E + (size_t)(m0 + 16 + r) * HD + da;
          v2f b0; b0.x = er0[0]; b0.y = er0[1];
          v2f b1; b1.x = er1[0]; b1.y = er1[1];
          e0 = wmma4f32(a, b0, e0);
          e1 = wmma4f32(a, b1, e1);
        }
        const int cc = r;
#pragma unroll
        for (int t8 = 0; t8 < 8; ++t8) {
          const int rr = t8 + 8 * half;
          const int t  = 15 - rr + cc;           // 0..30
          const int srcLane = (t & 15) + (half << 4);
          const float s0 = __shfl(e0[t8], srcLane, 32);
          const float s1 = __shfl(e1[t8], srcLane, 32);
          float srel = (t < 16) ? s0 : s1;
          if (cc + k0j > rr + q0) srel = 0.0f;   // k > q -> masked
          acc[j][t8] += srel;
        }
      }
    }

    // scaled store into LDS logits block
#pragma unroll
    for (int j = 0; j < 4; ++j) {
      const int kc = k0 + j * 16 + r;
#pragma unroll
      for (int t8 = 0; t8 < 8; ++t8) {
        const int rr = t8 + 8 * half;
        smem[rr * LSTR + kc] = acc[j][t8] * 0.125f;
      }
    }
  }
  __syncthreads();

  // ======== Phase 2: softmax per row in LDS (+ single HBM write) ========
#pragma unroll
  for (int rowi = 0; rowi < 2; ++rowi) {
    const int row = w * 2 + rowi;                // 8 waves x 2 rows = 16
    float* rp = smem + row * LSTR;

    float mx = -INFINITY;
    for (int i = lane; i < S; i += 32) mx = fmaxf(mx, rp[i]);
#pragma unroll
    for (int off = 16; off > 0; off >>= 1) mx = fmaxf(mx, __shfl_xor(mx, off, 32));

    float sum = 0.f;
    for (int i = lane; i < S; i += 32) {
      const float e = __expf(rp[i] - mx);
      rp[i] = e;
      sum += e;
    }
#pragma unroll
    for (int off = 16; off > 0; off >>= 1) sum += __shfl_xor(sum, off, 32);
    const float inv = 1.0f / sum;

    float* gout = attnW + ((size_t)bh * S + q0 + row) * S;
    for (int i = lane; i < S; i += 32) {
      const float a = rp[i] * inv;
      rp[i] = a;                                 // keep for phase 3
      gout[i] = a;                               // the only HBM write of attn
    }
  }
  __syncthreads();

  // ======== Phase 3: attnV(16 x 64) = attn(16x2048) @ V(2048x64) ========
  const int nt  = w & 3;                         // 16-col n-tile
  const int kh2 = w >> 2;                        // K half
  const int kbeg = kh2 * (S / 2);

  v8f acc = vzero8();
#pragma unroll 4
  for (int k = 0; k < S / 2; k += 4) {
    const int ka = kbeg + k + 2 * half;
    v2f a; a.x = smem[r * LSTR + ka]; a.y = smem[r * LSTR + ka + 1];
    const float* v0 = V + (size_t)ka * HD + nt * 16 + r;
    v2f b; b.x = v0[0]; b.y = v0[HD];
    acc = wmma4f32(a, b, acc);
  }

  float* spart = smem + 16 * LSTR;               // 4 x (16x16) staging
  if (kh2 == 1) {
#pragma unroll
    for (int t8 = 0; t8 < 8; ++t8)
      spart[nt * 256 + (t8 + 8 * half) * 16 + r] = acc[t8];
  }
  __syncthreads();
  if (kh2 == 0) {
    const int b  = bh >> 3, hh = bh & 7;
#pragma unroll
    for (int t8 = 0; t8 < 8; ++t8) {
      const int rr = t8 + 8 * half;
      const float val = acc[t8] + spart[nt * 256 + rr * 16 + r];
      attnV[((size_t)(b * S + q0 + rr)) * D + hh * HD + nt * 16 + r] = val;
    }
  }
}

// -------------------------------------------------------------------------
extern "C" void kernel_launch(void* const* d_in, const int* in_sizes, int n_in,
                              void* d_out, int out_size, void* d_ws, size_t ws_size,
                              hipStream_t stream)
{
  (void)in_sizes; (void)n_in; (void)out_size; (void)ws_size;

  const float* q  = (const float*)d_in[0];
  const float* k  = (const float*)d_in[1];
  const float* v  = (const float*)d_in[2];
  const float* E  = (const float*)d_in[3];
  const float* Wq = (const float*)d_in[4];
  const float* bq = (const float*)d_in[5];
  const float* Wk = (const float*)d_in[6];
  const float* bk = (const float*)d_in[7];
  const float* Wv = (const float*)d_in[8];
  const float* bv = (const float*)d_in[9];
  const float* Wo = (const float*)d_in[10];
  const float* bo = (const float*)d_in[11];

  float* out   = (float*)d_out;                         // (B,S,D)
  float* attnW = out + (size_t)BSZ * S * D;             // (B,H,S,S) output

  const size_t proj_elems = (size_t)BSZ * S * D;        // 2,097,152
  float* qh = (float*)d_ws;                             // (B,H,S,hd)
  float* kh = qh + proj_elems;
  float* vh = kh + proj_elems;
  float* aV = vh + proj_elems;                          // (B,S,D)

  // 1) input projections -> head layout
  gemm_proj_kernel<<<256, 256, 0, stream>>>(q, Wq, bq, qh, 0);
  gemm_proj_kernel<<<256, 256, 0, stream>>>(k, Wk, bk, kh, 0);
  gemm_proj_kernel<<<256, 256, 0, stream>>>(v, Wv, bv, vh, 0);

  // 2) fused logits + softmax + attn@V (attn matrix hits HBM exactly once)
  fused_attn_kernel<<<BSZ * H * (S / 16), 256, SMEM_BYTES, stream>>>(
      qh, kh, vh, E, attnW, aV);

  // 3) output projection
  gemm_proj_kernel<<<256, 256, 0, stream>>>(aV, Wo, bo, out, 1);
}